// IDWT3DTiny_88484916232343
// MI455X (gfx1250) — compile-verified
//
#include <hip/hip_runtime.h>
#include <cstdint>

// Inverse 3D Haar DWT with band_low = [1/sqrt2, 1/sqrt2] is exactly a 2x
// nearest-neighbor upsample per axis scaled by (1/sqrt2)^3:
//   out[2t+a, 2h+b, 2w+d, c] = x[t,h,w,c] * 2^-1.5   for a,b,d in {0,1}
//
// Shapes: x (16,128,128,64) f32  ->  out (32,256,256,64) f32
// Pure bandwidth problem (~604 MB HBM traffic => ~26 us at 23.3 TB/s).
// Strategy: stage each input (t,h) row (128*64 f32 = 32KB) into LDS via the
// CDNA5 async global->LDS path, then replicate x8 with 128-bit NT stores.

typedef float v4f __attribute__((ext_vector_type(4)));

#define SCALE 0.35355339059327373f   // 2^-1.5

__global__ __launch_bounds__(256) void idwt3_haar_upsample(
    const float* __restrict__ x, float* __restrict__ out) {
  // One input row (t,h,*,*) = 128 w * 64 c floats = 2048 v4f = 32 KB
  __shared__ v4f lds4[2048];

  const int tid = threadIdx.x;            // 0..255
  const int b   = blockIdx.x;             // 0..2047 = t*128 + h
  const int t   = b >> 7;
  const int h   = b & 127;

  // ---- Stage input row into LDS with async b128 copies (ASYNCcnt path) ----
  const v4f* in4 = (const v4f*)x + (uint64_t)b * 2048u;   // row base, v4f units
  const uint32_t lds_base = (uint32_t)(uintptr_t)(&lds4[0]); // low 32 bits of
                                                             // flat LDS ptr ==
                                                             // wave LDS offset
#pragma unroll
  for (int j = 0; j < 8; ++j) {
    const int e = j * 256 + tid;                            // v4f idx in row
    const uint64_t gaddr = (uint64_t)(uintptr_t)(in4 + e);  // global byte addr
    const uint32_t laddr = lds_base + (uint32_t)e * 16u;    // LDS byte addr
    // VDST = LDS address VGPR, VADDR = 64-bit global address, GV mode.
    asm volatile("global_load_async_to_lds_b128 %0, %1, off"
                 :: "v"(laddr), "v"(gaddr)
                 : "memory");
  }
  asm volatile("s_wait_asynccnt 0" ::: "memory");
  __syncthreads();

  // ---- Replicate x8: 4 output rows, each W duplicated within the row ----
  // Output row stride (one H step) = 256 W * 64 c = 16384 floats = 4096 v4f.
  v4f* o00 = (v4f*)out + ((uint64_t)(2 * t) * 256u + (uint64_t)(2 * h)) * 4096u;
  v4f* o01 = o00 + 4096u;                          // H = 2h+1
  v4f* o10 = o00 + (uint64_t)256u * 4096u;         // T = 2t+1
  v4f* o11 = o10 + 4096u;

#pragma unroll 4
  for (int i = 0; i < 16; ++i) {
    const int idx = i * 256 + tid;        // 0..4095 : output v4f within a row
    const int W   = idx >> 4;             // 0..255  (16 v4f of c per W)
    const int cq  = idx & 15;             // c-quad index
    // Input element at w = W>>1. Lanes 16..31 of a wave broadcast-read the
    // same 256B as lanes 0..15 (W pair shares data): conflict-free.
    v4f v = lds4[(W >> 1) * 16 + cq] * SCALE;
    // Non-temporal 128-bit stores: output is write-once, keep input in L2.
    __builtin_nontemporal_store(v, o00 + idx);
    __builtin_nontemporal_store(v, o01 + idx);
    __builtin_nontemporal_store(v, o10 + idx);
    __builtin_nontemporal_store(v, o11 + idx);
  }
}

extern "C" void kernel_launch(void* const* d_in, const int* in_sizes, int n_in,
                              void* d_out, int out_size, void* d_ws, size_t ws_size,
                              hipStream_t stream) {
  const float* x = (const float*)d_in[0];   // (16,128,128,64) f32
  float* out = (float*)d_out;               // (32,256,256,64) f32
  (void)in_sizes; (void)n_in; (void)out_size; (void)d_ws; (void)ws_size;

  dim3 grid(16 * 128);   // one block per input (t,h) row
  dim3 block(256);       // 8 wave32
  hipLaunchKernelGGL(idwt3_haar_upsample, grid, block, 0, stream, x, out);
}